// PtModule_73667279061061
// MI455X (gfx1250) — compile-verified
//
#include <hip/hip_runtime.h>

// Native 128-bit vector type so __builtin_nontemporal_load/store accept it
// (HIP's float4 is a class type; clang's nontemporal builtins want a real
// scalar/vector type).
typedef float v4f __attribute__((ext_vector_type(4)));

static constexpr int BLOCK = 256;   // 8 wave32s per workgroup
static constexpr int GRID  = 2048;  // plenty of blocks to saturate all WGPs

// wave32 butterfly reduction (gfx1250 is wave32-only; hardcoding width 32 is
// correct here and lowers to the LDS-permute path, no LDS memory touched).
__device__ __forceinline__ float wave_reduce_add(float v) {
    v += __shfl_xor(v, 16, 32);
    v += __shfl_xor(v, 8, 32);
    v += __shfl_xor(v, 4, 32);
    v += __shfl_xor(v, 2, 32);
    v += __shfl_xor(v, 1, 32);
    return v;
}

__device__ __forceinline__ float block_reduce_add(float v, float* smem) {
    const int lane = threadIdx.x & 31;
    const int wave = threadIdx.x >> 5;
    v = wave_reduce_add(v);
    if (lane == 0) smem[wave] = v;
    __syncthreads();
    if (wave == 0) {
        float w = (lane < (BLOCK >> 5)) ? smem[lane] : 0.0f;
        w = wave_reduce_add(w);
        return w;  // valid in lane 0
    }
    return 0.0f;
}

// Single streaming pass: read x once (NT b128), emit four fused elementwise
// outputs (NT b128 stores), and accumulate a per-block partial sum.
__global__ __launch_bounds__(BLOCK) void fused_stream_kernel(
    const float* __restrict__ x,
    float* __restrict__ out1,   // x + 1
    float* __restrict__ out2,   // x - 1
    float* __restrict__ out3,   // x * 2
    float* __restrict__ out4,   // x / 2  (== x * 0.5, exact in binary FP)
    float* __restrict__ partials,
    long long n)                 // total scalar elements
{
    __shared__ float smem[BLOCK >> 5];

    const v4f* __restrict__ xv  = (const v4f*)x;
    v4f* __restrict__ o1        = (v4f*)out1;
    v4f* __restrict__ o2        = (v4f*)out2;
    v4f* __restrict__ o3        = (v4f*)out3;
    v4f* __restrict__ o4        = (v4f*)out4;

    const long long n4     = n >> 2;
    const long long stride = (long long)gridDim.x * (long long)blockDim.x;
    long long i = (long long)blockIdx.x * blockDim.x + threadIdx.x;

    float acc = 0.0f;
    for (; i < n4; i += stride) {
        // Prefetch one grid-stride ahead (global_prefetch_b8, speculative).
        if (i + stride < n4)
            __builtin_prefetch((const void*)&xv[i + stride], 0, 0);

        v4f v = __builtin_nontemporal_load(&xv[i]);

        v4f a = v + 1.0f;
        v4f b = v - 1.0f;
        v4f c = v * 2.0f;
        v4f d = v * 0.5f;

        __builtin_nontemporal_store(a, &o1[i]);
        __builtin_nontemporal_store(b, &o2[i]);
        __builtin_nontemporal_store(c, &o3[i]);
        __builtin_nontemporal_store(d, &o4[i]);

        acc += (v.x + v.y) + (v.z + v.w);
    }

    // Scalar tail (n is a multiple of 4 for this shape; kept for robustness).
    for (long long j = (n4 << 2) + (long long)blockIdx.x * blockDim.x + threadIdx.x;
         j < n; j += stride) {
        float v = x[j];
        out1[j] = v + 1.0f;
        out2[j] = v - 1.0f;
        out3[j] = v * 2.0f;
        out4[j] = v * 0.5f;
        acc += v;
    }

    float bsum = block_reduce_add(acc, smem);
    if (threadIdx.x == 0) partials[blockIdx.x] = bsum;
}

// Deterministic finalize: fixed-shape tree over the per-block partials.
__global__ __launch_bounds__(BLOCK) void finalize_sum_kernel(
    const float* __restrict__ partials, float* __restrict__ out_sum, int n)
{
    __shared__ float smem[BLOCK >> 5];
    float acc = 0.0f;
    for (int i = threadIdx.x; i < n; i += BLOCK) acc += partials[i];
    float s = block_reduce_add(acc, smem);
    if (threadIdx.x == 0) *out_sum = s;
}

extern "C" void kernel_launch(void* const* d_in, const int* in_sizes, int n_in,
                              void* d_out, int out_size, void* d_ws, size_t ws_size,
                              hipStream_t stream) {
    (void)n_in; (void)out_size; (void)ws_size;

    const long long n = (long long)in_sizes[0];  // 4096*16384 = 67,108,864
    const float* x = (const float*)d_in[0];
    float* out = (float*)d_out;

    // Output tuple laid out flat in return order: x1, x2, x3, x4, sum.
    float* o1 = out;
    float* o2 = out + n;
    float* o3 = out + 2 * n;
    float* o4 = out + 3 * n;
    float* sum_out = out + 4 * n;

    float* partials = (float*)d_ws;  // GRID floats of scratch (8 KiB)

    fused_stream_kernel<<<GRID, BLOCK, 0, stream>>>(x, o1, o2, o3, o4,
                                                    partials, n);
    finalize_sum_kernel<<<1, BLOCK, 0, stream>>>(partials, sum_out, GRID);
}